// LoRALinear_41446434406516
// MI455X (gfx1250) — compile-verified
//
#include <hip/hip_runtime.h>
#include <hip/hip_bf16.h>

typedef __attribute__((ext_vector_type(2))) float v2f;
typedef __attribute__((ext_vector_type(8))) float v8f;

#define K_DIM 4096   // inner dim of x@A
#define N_DIM 4096   // output columns
#define M_TOT 16384  // 4 * 4096 rows, x flattened [16384, 4096]

static __device__ __forceinline__ v8f wmma_f32_16x16x4(v2f a, v2f b, v8f c) {
#if __has_builtin(__builtin_amdgcn_wmma_f32_16x16x4_f32)
  return __builtin_amdgcn_wmma_f32_16x16x4_f32(false, a, false, b, (short)0, c,
                                               false, false);
#else
  asm volatile("v_wmma_f32_16x16x4_f32 %0, %1, %2, %0"
               : "+v"(c)
               : "v"(a), "v"(b));
  return c;
#endif
}

// One workgroup = 8 wave32 = 256 threads, owns 16 rows of x / out.
// Phase 1: T(16x16) = Xslab(16x4096) @ A(4096x16)   (waves split K)
// Phase 2: Out(16x4096) = T(16x16) @ B(16x4096)     (waves split N)
__global__ __launch_bounds__(256) void lora_fused_kernel(
    const float* __restrict__ x, const float* __restrict__ A,
    const float* __restrict__ B, float* __restrict__ out) {
  __shared__ float lds_part[8][256];  // per-wave 16x16 partials (WMMA layout)
  __shared__ float lds_T[16][16];     // reduced intermediate T

  const int tid = threadIdx.x;
  const int wave = tid >> 5;       // 0..7 (wave32)
  const int lane = tid & 31;
  const int halfSel = lane >> 4;   // 0 = lanes 0-15, 1 = lanes 16-31
  const int lane16 = lane & 15;
  const long rowBase = (long)blockIdx.x * 16;

  // ------------------- Phase 1: T = Xslab @ A -------------------
  // A-operand (16x4 f32): lanes 0-15 hold row M=lane, K = k0,k0+1
  //                       lanes 16-31 hold row M=lane-16, K = k0+2,k0+3
  // B-operand (4x16 f32): VGPR0 = row k0 (+2 for hi lanes), col = lane16
  v8f acc = {};
  const float* xrow = x + (rowBase + lane16) * (long)K_DIM;
  const int kBegin = wave * (K_DIM / 8);

#pragma unroll 4
  for (int k0 = kBegin; k0 < kBegin + (K_DIM / 8); k0 += 4) {
    const int kk = k0 + halfSel * 2;
    v2f ax = *(const v2f*)(xrow + kk);  // global_load_b64, line-reuse over k
    v2f ab;
    ab.x = A[(long)(kk + 0) * 16 + lane16];  // LoRA A is [4096][16], L2-resident
    ab.y = A[(long)(kk + 1) * 16 + lane16];
    acc = wmma_f32_16x16x4(ax, ab, acc);
  }

  // Spill per-wave partial accumulators to LDS in WMMA C/D layout:
  // element i of lane L  ->  T[i + (L>=16)*8][L&15]
#pragma unroll
  for (int i = 0; i < 8; ++i) lds_part[wave][i * 32 + lane] = acc[i];
  __syncthreads();

  // Cross-wave reduction: thread tid owns T[m][n], m = tid/16, n = tid%16
  {
    const int m = tid >> 4, n = tid & 15;
    float s = 0.f;
#pragma unroll
    for (int w = 0; w < 8; ++w)
      s += lds_part[w][(m & 7) * 32 + n + (m >> 3) * 16];
    lds_T[m][n] = s;
  }
  __syncthreads();

  // ------------------- Phase 2: Out = T @ B -------------------
  // Load the four 16x4 T fragments once (reused for every N tile).
  v2f tf[4];
#pragma unroll
  for (int j = 0; j < 4; ++j) {
    const int kk = j * 4 + halfSel * 2;
    tf[j].x = lds_T[lane16][kk + 0];
    tf[j].y = lds_T[lane16][kk + 1];
  }

  // 256 N-tiles of 16 columns; wave w takes tiles w, w+8, w+16, ...
  for (int t = 0; t < 32; ++t) {
    const long col0 = (long)(t * 8 + wave) * 16;
    v8f c = {};
#pragma unroll
    for (int j = 0; j < 4; ++j) {
      const int kk = j * 4 + halfSel * 2;
      v2f bf;
      bf.x = B[(long)(kk + 0) * N_DIM + col0 + lane16];  // B L2-resident
      bf.y = B[(long)(kk + 1) * N_DIM + col0 + lane16];
      c = wmma_f32_16x16x4(tf[j], bf, c);
    }
    // D layout: VGPR i -> row (i + halfSel*8), col = lane16. 64B segments.
#pragma unroll
    for (int i = 0; i < 8; ++i) {
      out[(rowBase + i + halfSel * 8) * (long)N_DIM + col0 + lane16] = c[i];
    }
  }
}

extern "C" void kernel_launch(void* const* d_in, const int* in_sizes, int n_in,
                              void* d_out, int out_size, void* d_ws,
                              size_t ws_size, hipStream_t stream) {
  const float* x = (const float*)d_in[0];  // [4,4096,4096] -> [16384,4096]
  const float* A = (const float*)d_in[1];  // [4096,16]
  const float* B = (const float*)d_in[2];  // [16,4096]
  float* out = (float*)d_out;              // [16384,4096]
  (void)in_sizes; (void)n_in; (void)out_size; (void)d_ws; (void)ws_size;

  dim3 grid(M_TOT / 16);  // 1024 workgroups, one 16-row slab each
  dim3 block(256);        // 8 wave32
  lora_fused_kernel<<<grid, block, 0, stream>>>(x, A, B, out);
}